// ContrastiveLoss_70712341561919
// MI455X (gfx1250) — compile-verified
//
#include <hip/hip_runtime.h>
#include <hip/hip_bf16.h>
#include <math.h>

// ---------------------------------------------------------------------------
// ContrastiveLoss for MI455X (gfx1250, wave32).
// B=32, C=256, D=64 -> P=2080 (multiple of 16), T=64, K=3.
// HBM-bound problem: keep fp32, use V_WMMA_F32_16X16X4_F32 for the GEMMs,
// fuse normalization into the WMMA B-loads and the exp/mask log-sum-exp
// reductions onto the accumulators. A-operands staged in LDS (bank-padded),
// all hot-loop loads unconditional so the compiler can keep them in flight.
// ---------------------------------------------------------------------------

typedef __attribute__((ext_vector_type(2))) float v2f;
typedef __attribute__((ext_vector_type(8))) float v8f;

#define K_TOP   3
#define INV_TV  10.0f   // 1 / 0.1
#define INV_TQ  10.0f   // 1 / 0.1
#define NEG_IOU 0.5f
#define MAXC    256
#define LDS_STRIDE (MAXC + 2)   // +2 dwords: de-conflict LDS banks, keep 8B align

__device__ __forceinline__ float wave_sum32(float v) {
    v += __shfl_xor(v, 16);
    v += __shfl_xor(v, 8);
    v += __shfl_xor(v, 4);
    v += __shfl_xor(v, 2);
    v += __shfl_xor(v, 1);
    return v;
}
// sum over the 16 lanes of each half-wave (lanes 0-15 / 16-31 kept separate)
__device__ __forceinline__ float half_sum16(float v) {
    v += __shfl_xor(v, 1);
    v += __shfl_xor(v, 2);
    v += __shfl_xor(v, 4);
    v += __shfl_xor(v, 8);
    return v;
}

// --------------------------------------------------------------------------
// 1) init: triu p -> (r,c) flat offset map, prefix sums, zero accumulators
// --------------------------------------------------------------------------
__global__ void init_kernel(int* __restrict__ poff, int* __restrict__ tgt_start,
                            int* __restrict__ row_start,
                            float* __restrict__ NS_q, float* __restrict__ negsum_row,
                            float* __restrict__ acc, const int* __restrict__ nt,
                            int B, int D, int P, int totalRows) {
    int tid = threadIdx.x;
    for (int p = tid; p < P; p += blockDim.x) {
        int r = 0, rs = 0;
        while (r + 1 < D) {
            int next = rs + (D - r);
            if (next > p) break;
            rs = next; ++r;
        }
        int c = r + (p - rs);
        poff[p] = r * D + c;
    }
    for (int i = tid; i < B; i += blockDim.x) NS_q[i] = 0.f;
    for (int i = tid; i < totalRows; i += blockDim.x) negsum_row[i] = 0.f;
    if (tid < 8) acc[tid] = 0.f;
    if (tid == 0) {
        int ts = 0, rs2 = 0;
        for (int v = 0; v < B; ++v) {
            tgt_start[v] = ts;
            row_start[v] = rs2;
            ts += nt[v];
            rs2 += nt[v] * K_TOP;
        }
    }
}

// --------------------------------------------------------------------------
// 2) normalize query_feats (B rows) and sents_feats (T rows); one wave/row
// --------------------------------------------------------------------------
__global__ void normalize_rows(const float* __restrict__ q, const float* __restrict__ s,
                               float* __restrict__ qfn, float* __restrict__ sfn,
                               int B, int T, int C) {
    int wave = blockIdx.x * (blockDim.x >> 5) + (threadIdx.x >> 5);
    int lane = threadIdx.x & 31;
    if (wave >= B + T) return;
    const float* src = (wave < B) ? q + (size_t)wave * C : s + (size_t)(wave - B) * C;
    float*       dst = (wave < B) ? qfn + (size_t)wave * C : sfn + (size_t)(wave - B) * C;
    float ss = 0.f;
    for (int c = lane; c < C; c += 32) { float x = src[c]; ss += x * x; }
    ss = wave_sum32(ss);
    float rn = 1.f / fmaxf(sqrtf(ss), 1e-12f);
    for (int c = lane; c < C; c += 32) dst[c] = src[c] * rn;
}

// --------------------------------------------------------------------------
// 3) per-proposal reciprocal norms over channels (coalesced across p)
// --------------------------------------------------------------------------
__global__ void rnorm_kernel(const float* __restrict__ vf, const int* __restrict__ poff,
                             float* __restrict__ rnorm, int C, int D, int P) {
    int b = blockIdx.y;
    int p = blockIdx.x * blockDim.x + threadIdx.x;
    if (p >= P) return;
    int DD = D * D;
    const float* base = vf + (size_t)b * C * DD + poff[p];
    float ss = 0.f;
    for (int c = 0; c < C; ++c) { float x = base[(size_t)c * DD]; ss += x * x; }
    rnorm[(size_t)b * P + p] = 1.f / fmaxf(sqrtf(ss), 1e-12f);
}

// --------------------------------------------------------------------------
// 4) top-K=3 of iou2ds per target (tiny: T threads scan P)
// --------------------------------------------------------------------------
__global__ void topk_kernel(const float* __restrict__ iou2ds, const int* __restrict__ poff,
                            int* __restrict__ topk, int T, int P, int D) {
    int t = blockIdx.x * blockDim.x + threadIdx.x;
    if (t >= T) return;
    int DD = D * D;
    const float* row = iou2ds + (size_t)t * DD;
    float v0 = -1e30f, v1 = -1e30f, v2 = -1e30f;
    int i0 = 0, i1 = 0, i2 = 0;
    for (int p = 0; p < P; ++p) {
        float x = row[poff[p]];
        if (x > v0)      { v2 = v1; i2 = i1; v1 = v0; i1 = i0; v0 = x; i0 = p; }
        else if (x > v1) { v2 = v1; i2 = i1; v1 = x;  i1 = p; }
        else if (x > v2) { v2 = x;  i2 = p; }
    }
    topk[t * K_TOP + 0] = i0;
    topk[t * K_TOP + 1] = i1;
    topk[t * K_TOP + 2] = i2;
}

// --------------------------------------------------------------------------
// 5) gather normalized top-k proposal features -> tvf (totalRows x C)
// --------------------------------------------------------------------------
__global__ void gather_tv(const float* __restrict__ vf, const float* __restrict__ rnorm,
                          const int* __restrict__ poff, const int* __restrict__ topk,
                          const int* __restrict__ scat, float* __restrict__ tvf,
                          int C, int D, int P, int totalRows) {
    int idx = blockIdx.x * blockDim.x + threadIdx.x;
    if (idx >= totalRows * C) return;
    int row = idx / C, c = idx - row * C;
    int t = row / K_TOP;
    int v = scat[t];
    int p = topk[row];
    int DD = D * D;
    tvf[idx] = vf[((size_t)v * C + c) * DD + poff[p]] * rnorm[(size_t)v * P + p];
}

// --------------------------------------------------------------------------
// 6) inter-video loss: one wave per (t,k) row; lane b computes tv . q_b
// --------------------------------------------------------------------------
__global__ void inter_video_kernel(const float* __restrict__ tvf, const float* __restrict__ qfn,
                                   const int* __restrict__ scat, float* __restrict__ pos_tk,
                                   float* __restrict__ acc, int B, int C, int totalRows) {
    int wave = blockIdx.x * (blockDim.x >> 5) + (threadIdx.x >> 5);
    int lane = threadIdx.x & 31;
    if (wave >= totalRows) return;
    int t = wave / K_TOP;
    int v = scat[t];
    float d = 0.f;
    if (lane < B) {
        const float* tr = tvf + (size_t)wave * C;
        const float* qr = qfn + (size_t)lane * C;
        for (int c = 0; c < C; ++c) d += tr[c] * qr[c];
    }
    float pos = __shfl(d, v);
    float e = (lane < B && lane != v) ? __expf(d * INV_TV) : 0.f;
    e = wave_sum32(e);
    if (lane == 0) {
        pos_tk[wave] = pos;
        float pt = pos * INV_TV;
        atomicAdd(&acc[0], -(pt - logf(__expf(pt) + e)));
    }
}

// --------------------------------------------------------------------------
// 7) FUSED WMMA kernel (V_WMMA_F32_16X16X4_F32).
//    grid = (ceil(ptiles/4), B): one workgroup per video x 4 proposal tiles.
//    A-operands (qfn + this video's tvf rows) staged zero-padded in LDS, so
//    the hot loop has only unconditional loads:
//      2x global_load_b32 (streaming video_feats, the unavoidable traffic)
//      3x ds_load_b64     (A operands, bank-conflict-free via padded stride)
//      3x v_wmma_f32_16x16x4_f32
//    Produces NS_q[m] (inter-query negatives) and negsum_row (intra-video).
// --------------------------------------------------------------------------
__global__ void gemm_negsums(const float* __restrict__ vfeat,
                             const float* __restrict__ iou2d,
                             const float* __restrict__ iou2ds,
                             const float* __restrict__ qfn,
                             const float* __restrict__ tvf,
                             const float* __restrict__ rnorm,
                             const int* __restrict__ poff,
                             const int* __restrict__ tgt_start,
                             const int* __restrict__ row_start,
                             const int* __restrict__ nt,
                             float* __restrict__ NS_q,
                             float* __restrict__ negsum_row,
                             int B, int C, int D, int P, int ptiles) {
    __shared__ float lds_q[32 * LDS_STRIDE];   // all queries (rows >= B zeroed)
    __shared__ float lds_t[16 * LDS_STRIDE];   // this video's tvf rows (zeroed pad)

    const int bcol = blockIdx.y;
    const int tid = threadIdx.x;
    const int rs = row_start[bcol];
    int cnt = nt[bcol] * K_TOP;
    if (cnt > 16) cnt = 16;
    const int ts = tgt_start[bcol];
    const int DD = D * D;

    // ---- stage A operands in LDS, zero-padded ----
    for (int i = tid; i < 32 * MAXC; i += blockDim.x) {
        int r = i / MAXC, c = i - r * MAXC;
        lds_q[r * LDS_STRIDE + c] = (r < B && c < C) ? qfn[(size_t)r * C + c] : 0.f;
    }
    for (int i = tid; i < 16 * MAXC; i += blockDim.x) {
        int r = i / MAXC, c = i - r * MAXC;
        lds_t[r * LDS_STRIDE + c] = (r < cnt && c < C) ? tvf[(size_t)(rs + r) * C + c] : 0.f;
    }
    __syncthreads();

    const int lane = tid & 31;
    const int tile = blockIdx.x * (blockDim.x >> 5) + (tid >> 5);
    if (tile >= ptiles) return;
    const int p0 = tile * 16;

    const int n = lane & 15;
    const int p = p0 + n;
    const bool pvalid = (p < P);
    const int pc = pvalid ? p : (P - 1);          // clamped address, no branchy load
    const int po = poff[pc];
    float rn = rnorm[(size_t)bcol * P + pc];
    if (!pvalid) rn = 0.f;                        // zeroes the whole B column

    const int khalf = (lane >> 4) * 2;            // K sub-pair of this half-wave
    const int ma = lane & 15;                     // A row within tile

    const float* bbase = vfeat + (size_t)bcol * C * DD + po;
    const float* q0 = &lds_q[ma * LDS_STRIDE];
    const float* q1 = &lds_q[(ma + 16) * LDS_STRIDE];
    const float* t2 = &lds_t[ma * LDS_STRIDE];

    v8f c0 = {}; v8f c1 = {}; v8f c2 = {};
#pragma unroll 4
    for (int k0 = 0; k0 < C; k0 += 4) {
        int k = k0 + khalf;
        v2f a0 = *(const v2f*)(q0 + k);           // ds_load_b64, unconditional
        v2f a1 = *(const v2f*)(q1 + k);
        v2f a2 = *(const v2f*)(t2 + k);
        v2f bv;
        bv.x = bbase[(size_t)k * DD] * rn;        // streaming global loads
        bv.y = bbase[(size_t)(k + 1) * DD] * rn;
        c0 = __builtin_amdgcn_wmma_f32_16x16x4_f32(false, a0, false, bv, (short)0, c0, false, false);
        c1 = __builtin_amdgcn_wmma_f32_16x16x4_f32(false, a1, false, bv, (short)0, c1, false, false);
        c2 = __builtin_amdgcn_wmma_f32_16x16x4_f32(false, a2, false, bv, (short)0, c2, false, false);
    }

    // masks depending only on this lane's proposal p
    const float iouv = pvalid ? iou2d[(size_t)bcol * DD + po] : 0.f;
    const bool iou_hi = (iouv > NEG_IOU);
    const int mofs = (lane >> 4) << 3;            // D-layout: lanes>=16 hold m+8

#pragma unroll
    for (int i = 0; i < 8; ++i) {
        int mloc = i + mofs;
        // inter-query negatives, query rows 0..15
        {
            int m = mloc;
            float e = (pvalid && m < B && !(m == bcol && iou_hi)) ? __expf(c0[i] * INV_TQ) : 0.f;
            e = half_sum16(e);
            if (n == 0 && m < B) atomicAdd(&NS_q[m], e);
        }
        // inter-query negatives, query rows 16..31
        {
            int m = mloc + 16;
            float e = (pvalid && m < B && !(m == bcol && iou_hi)) ? __expf(c1[i] * INV_TQ) : 0.f;
            e = half_sum16(e);
            if (n == 0 && m < B) atomicAdd(&NS_q[m], e);
        }
        // intra-video negatives for this video's tvf rows
        {
            bool mv = (mloc < cnt);
            float e = 0.f;
            if (pvalid && mv) {
                int t = ts + mloc / K_TOP;
                float iod = iou2ds[(size_t)t * DD + po];
                if (iod < NEG_IOU) e = __expf(c2[i] * INV_TV);
            }
            e = half_sum16(e);
            if (n == 0 && mv) atomicAdd(&negsum_row[rs + mloc], e);
        }
    }
}

// --------------------------------------------------------------------------
// 8) inter-query loss: reuse pos_tk with NS_q
// --------------------------------------------------------------------------
__global__ void inter_query_fin(const float* __restrict__ pos_tk, const float* __restrict__ NS_q,
                                const int* __restrict__ scat, float* __restrict__ acc,
                                int totalRows) {
    int i = blockIdx.x * blockDim.x + threadIdx.x;
    if (i >= totalRows) return;
    int v = scat[i / K_TOP];
    float pt = pos_tk[i] * INV_TQ;
    atomicAdd(&acc[1], -(pt - logf(__expf(pt) + NS_q[v])));
}

// --------------------------------------------------------------------------
// 9) intra-video loss over all (i,j) pairs within each video's nK rows
// --------------------------------------------------------------------------
__global__ void pairs_kernel(const float* __restrict__ tvf, const float* __restrict__ negsum_row,
                             const int* __restrict__ row_start, const int* __restrict__ nt,
                             float* __restrict__ acc, int C) {
    int v = blockIdx.x;
    int cnt = nt[v] * K_TOP;
    int rs = row_start[v];
    int npairs = cnt * cnt;
    for (int idx = threadIdx.x; idx < npairs; idx += blockDim.x) {
        int i = idx / cnt, j = idx - i * cnt;
        const float* ra = tvf + (size_t)(rs + i) * C;
        const float* rb = tvf + (size_t)(rs + j) * C;
        float pos = 0.f;
        for (int c = 0; c < C; ++c) pos += ra[c] * rb[c];
        float pt = pos * INV_TV;
        atomicAdd(&acc[2], -(pt - logf(__expf(pt) + negsum_row[rs + i])));
        atomicAdd(&acc[3], 1.0f);
    }
}

// --------------------------------------------------------------------------
// 10) intra-query loss: one wave per target of a multi-target video
// --------------------------------------------------------------------------
__global__ void intra_query_kernel(const float* __restrict__ sfn, const float* __restrict__ qfn,
                                   const int* __restrict__ scat, const int* __restrict__ nt,
                                   float* __restrict__ acc, int B, int C, int T) {
    int wave = blockIdx.x * (blockDim.x >> 5) + (threadIdx.x >> 5);
    int lane = threadIdx.x & 31;
    if (wave >= T) return;
    int v = scat[wave];
    if (nt[v] <= 1) return;
    bool valid = (lane < B) && (nt[lane] > 1);
    float d = 0.f;
    if (valid) {
        const float* sr = sfn + (size_t)wave * C;
        const float* qr = qfn + (size_t)lane * C;
        for (int c = 0; c < C; ++c) d += sr[c] * qr[c];
    }
    float pos = __shfl(d, v);
    float e = (valid && lane != v) ? __expf(d * INV_TQ) : 0.f;
    e = wave_sum32(e);
    if (lane == 0) {
        float pt = pos * INV_TQ;
        atomicAdd(&acc[4], -(pt - logf(__expf(pt) + e)));
        atomicAdd(&acc[5], 1.0f);
    }
}

// --------------------------------------------------------------------------
// 11) finalize means -> out[0..3]
// --------------------------------------------------------------------------
__global__ void finalize_kernel(const float* __restrict__ acc, float* __restrict__ out,
                                float invTK) {
    if (threadIdx.x == 0 && blockIdx.x == 0) {
        out[0] = acc[0] * invTK;
        out[1] = acc[1] * invTK;
        out[2] = (acc[3] > 0.f) ? acc[2] / acc[3] : 0.f;
        out[3] = (acc[5] > 0.f) ? acc[4] / acc[5] : 0.f;
    }
}

// ---------------------------------------------------------------------------
extern "C" void kernel_launch(void* const* d_in, const int* in_sizes, int n_in,
                              void* d_out, int out_size, void* d_ws, size_t ws_size,
                              hipStream_t stream) {
    const float* vfeat  = (const float*)d_in[0];  // (B,C,D,D)
    const float* query  = (const float*)d_in[1];  // (B,C)
    const float* sents  = (const float*)d_in[2];  // (T,C)
    const float* iou2d  = (const float*)d_in[3];  // (B,D,D)
    const float* iou2ds = (const float*)d_in[4];  // (T,D,D)
    const int*   nt     = (const int*)d_in[5];    // (B,)
    const int*   scat   = (const int*)d_in[6];    // (T,)
    float* out = (float*)d_out;

    const int T = in_sizes[6];
    const int C = in_sizes[2] / T;
    const int B = in_sizes[1] / C;
    const int DD = in_sizes[3] / B;
    int D = 1;
    while ((D + 1) * (D + 1) <= DD) ++D;        // D = sqrt(DD)
    const int P = D * (D + 1) / 2;              // 2080 for D=64 (multiple of 16)
    const int totalRows = T * K_TOP;
    const int ptiles = (P + 15) / 16;

    // ---- workspace carve-out (256B aligned slices) ----
    char* ws = (char*)d_ws;
    size_t off = 0;
    auto alloc = [&](size_t bytes) {
        char* p = ws + off;
        off += (bytes + 255) & ~(size_t)255;
        return p;
    };
    float* rnorm      = (float*)alloc(sizeof(float) * (size_t)B * P);
    int*   poff       = (int*)  alloc(sizeof(int) * P);
    float* qfn        = (float*)alloc(sizeof(float) * (size_t)B * C);
    float* sfn        = (float*)alloc(sizeof(float) * (size_t)T * C);
    int*   topk       = (int*)  alloc(sizeof(int) * totalRows);
    float* tvf        = (float*)alloc(sizeof(float) * (size_t)totalRows * C);
    int*   tgt_start  = (int*)  alloc(sizeof(int) * B);
    int*   row_start  = (int*)  alloc(sizeof(int) * B);
    float* NS_q       = (float*)alloc(sizeof(float) * B);
    float* negsum_row = (float*)alloc(sizeof(float) * totalRows);
    float* pos_tk     = (float*)alloc(sizeof(float) * totalRows);
    float* acc        = (float*)alloc(sizeof(float) * 8);
    (void)ws_size; (void)n_in; (void)out_size;

    // 1) init maps / prefixes / accumulators
    init_kernel<<<1, 256, 0, stream>>>(poff, tgt_start, row_start, NS_q, negsum_row,
                                       acc, nt, B, D, P, totalRows);
    // 2) normalize query + sentence rows (one wave32 per row)
    {
        int waves = B + T;
        normalize_rows<<<(waves + 7) / 8, 256, 0, stream>>>(query, sents, qfn, sfn, B, T, C);
    }
    // 3) reciprocal norms of all proposals
    {
        dim3 g((P + 255) / 256, B);
        rnorm_kernel<<<g, 256, 0, stream>>>(vfeat, poff, rnorm, C, D, P);
    }
    // 4) top-K proposals per target
    topk_kernel<<<(T + 63) / 64, 64, 0, stream>>>(iou2ds, poff, topk, T, P, D);
    // 5) gather normalized top-k features
    gather_tv<<<(totalRows * C + 255) / 256, 256, 0, stream>>>(
        vfeat, rnorm, poff, topk, scat, tvf, C, D, P, totalRows);
    // 6) inter-video loss (+ store pos)
    inter_video_kernel<<<(totalRows + 7) / 8, 256, 0, stream>>>(
        tvf, qfn, scat, pos_tk, acc, B, C, totalRows);
    // 7) fused WMMA GEMM: inter-query + intra-video negative sums
    {
        dim3 g((ptiles + 3) / 4, B);            // 33 x 32 blocks of 4 waves
        gemm_negsums<<<g, 128, 0, stream>>>(
            vfeat, iou2d, iou2ds, qfn, tvf, rnorm, poff, tgt_start, row_start, nt,
            NS_q, negsum_row, B, C, D, P, ptiles);
    }
    // 8) inter-query loss
    inter_query_fin<<<(totalRows + 255) / 256, 256, 0, stream>>>(
        pos_tk, NS_q, scat, acc, totalRows);
    // 9) intra-video pairwise loss
    pairs_kernel<<<B, 256, 0, stream>>>(tvf, negsum_row, row_start, nt, acc, C);
    // 10) intra-query loss
    intra_query_kernel<<<(T + 7) / 8, 256, 0, stream>>>(sfn, qfn, scat, nt, acc, B, C, T);
    // 11) finalize
    finalize_kernel<<<1, 32, 0, stream>>>(acc, out, 1.0f / (float)(T * K_TOP));
}